// GNN_758
// MI455X (gfx1250) — compile-verified
//
#include <hip/hip_runtime.h>

typedef __attribute__((ext_vector_type(2))) float v2f;
typedef __attribute__((ext_vector_type(8))) float v8f;

#define DD 128

// ---------------------------------------------------------------------------
// zero a float buffer (n must be a multiple of 4)
// ---------------------------------------------------------------------------
__global__ void zero_f32(float* __restrict__ p, long n) {
  long i4 = ((long)blockIdx.x * blockDim.x + threadIdx.x) * 4;
  if (i4 < n) {
    float4 z = make_float4(0.f, 0.f, 0.f, 0.f);
    *(float4*)(p + i4) = z;
  }
}

// ---------------------------------------------------------------------------
// out[n, 128] = A[n, 128] @ B[128, 128]  (+ addsrc + bias optionally)
// fp32 WMMA 16x16x4. 128 threads = 4 waves; block covers 128 rows; each wave
// owns TWO 16-row strips (32 rows), so every B fragment feeds 2 WMMAs.
//
// LDS layout (B FIRST so all B-load immediates fit the 16-bit DS offset):
//   floats: Bev [0,8192) | pad 32 | Bod [8224,16416) | pad 32 | Aps [16448,32832)
//   B pairs {B[2kp][col], B[2kp+1][col]}: even-kp region at 0, odd-kp region
//   at +8224 floats (bank shift 32) -> the two lane halves of one ds_load_b64
//   cover disjoint bank halves; kp>>1 indexing gives one shared immediate
//   (1024*i + 128*ct bytes, max 32640) per load.
//   Aps transposed pairs [strip s][kp][r]: bank = (32*kp + 2r)%64 -> even-kp
//   half-wave banks 0..31, odd-kp half-wave banks 32..63 (base ≡ 0 mod 64).
//
// Fragment layouts per CDNA5 ISA 7.12.2:
//   A 16x4 fp32:  lanes 0-15 -> M=lane, v0=K0 v1=K1 ; lanes 16-31 -> v0=K2 v1=K3
//   B 4x16 fp32:  lanes 0-15 -> N=lane, v0=K0 v1=K1 ; lanes 16-31 -> v0=K2 v1=K3
//   C/D 16x16:    vgpr j: lanes 0-15 -> M=j, N=lane ; lanes 16-31 -> M=j+8
// ---------------------------------------------------------------------------
__global__ __launch_bounds__(128)
void gemm128_wmma(const float* __restrict__ A, const float* __restrict__ B,
                  const float* __restrict__ addsrc, const float* __restrict__ bias,
                  float* __restrict__ out, int nrows) {
  __shared__ __align__(16) float lds[32832];
  float* Bev = lds;                 // even-kp B pairs
  float* Bod = lds + 8224;          // odd-kp B pairs (+32-float bank shift)
  float* Aps = lds + 16448;         // transposed A pairs (base multiple of 64)

  const int tid  = threadIdx.x;
  const int wave = tid >> 5;    // 0..3
  const int lane = tid & 31;
  const int half = lane >> 4;   // which K-pair of the 4-wide K slice this lane holds
  const int l16  = lane & 15;
  const int blockRow = blockIdx.x * 128;

  // ---- stage B pairs into even/odd-kp regions: 16 tasks/thread ------------
#pragma unroll
  for (int i = 0; i < 16; ++i) {
    const int task = i * 128 + tid;          // 0..2047
    const int kp   = task >> 5;              // 0..63
    const int c4   = (task & 31) * 4;
    const float4 r0 = *(const float4*)(B + (size_t)(2 * kp) * DD + c4);
    const float4 r1 = *(const float4*)(B + (size_t)(2 * kp + 1) * DD + c4);
    float* dp = ((kp & 1) ? Bod : Bev) + ((size_t)(kp >> 1) * 128 + c4) * 2;
    v2f p;
    p.x = r0.x; p.y = r1.x; *(v2f*)(dp + 0) = p;
    p.x = r0.y; p.y = r1.y; *(v2f*)(dp + 2) = p;
    p.x = r0.z; p.y = r1.z; *(v2f*)(dp + 4) = p;
    p.x = r0.w; p.y = r1.w; *(v2f*)(dp + 6) = p;
  }

  // ---- stage A (128 rows, clamped) as transposed pairs: 32 tasks/thread ---
#pragma unroll
  for (int i = 0; i < 32; ++i) {
    const int task = i * 128 + tid;          // 0..4095
    const int rloc = task >> 5;              // 0..127 local row
    const int c4   = (task & 31) * 4;        // 0,4,...,124
    int row = blockRow + rloc;
    if (row > nrows - 1) row = nrows - 1;    // clamp: garbage rows are masked at store
    const float4 v = *(const float4*)(A + (size_t)row * DD + c4);
    const int s = rloc >> 4, r = rloc & 15;
    float* dp = Aps + ((size_t)s * 1024 + (c4 >> 1) * 16 + r) * 2;
    v2f p;
    p.x = v.x; p.y = v.y; *(v2f*)(dp)      = p;   // pair kp = c4/2
    p.x = v.z; p.y = v.w; *(v2f*)(dp + 32) = p;   // pair kp = c4/2 + 1
  }
  __syncthreads();

  v8f acc[2][8];
#pragma unroll
  for (int st = 0; st < 2; ++st)
#pragma unroll
    for (int t = 0; t < 8; ++t)
#pragma unroll
      for (int j = 0; j < 8; ++j) acc[st][t][j] = 0.f;

  // per-lane bases: runtime 'half' folds into the base, offsets are immediates
  const float* bB  = lds + (size_t)half * 8224 + 2 * l16;               // B region
  const float* bA0 = Aps + (size_t)(2 * wave) * 2048 + 32 * half + 2 * l16;
  const float* bA1 = bA0 + 2048;

  // ---- main loop: 2 A-loads + 8 B-loads per 16 WMMAs, all conflict-free ---
#pragma unroll
  for (int i = 0; i < 32; ++i) {             // K-slice of 4: kk = 4i + 2*half
    const v2f a0 = *(const v2f*)(bA0 + 64 * i);
    const v2f a1 = *(const v2f*)(bA1 + 64 * i);
#pragma unroll
    for (int ct = 0; ct < 8; ++ct) {
      const v2f b = *(const v2f*)(bB + 256 * i + 32 * ct);
      acc[0][ct] = __builtin_amdgcn_wmma_f32_16x16x4_f32(
          false, a0, false, b, (short)0, acc[0][ct], false, false);
      acc[1][ct] = __builtin_amdgcn_wmma_f32_16x16x4_f32(
          false, a1, false, b, (short)0, acc[1][ct], false, false);
    }
  }

  // ---- epilogue: optional residual add + bias, masked store ---------------
#pragma unroll
  for (int st = 0; st < 2; ++st) {
    const int stripRow = blockRow + (2 * wave + st) * 16 + half * 8;
#pragma unroll
    for (int ct = 0; ct < 8; ++ct) {
      const int col = ct * 16 + l16;
#pragma unroll
      for (int j = 0; j < 8; ++j) {
        const int row = stripRow + j;
        if (row < nrows) {
          float v = acc[st][ct][j];
          const size_t idx = (size_t)row * DD + col;
          if (addsrc) v += addsrc[idx];
          if (bias)   v += bias[col];
          out[idx] = v;
        }
      }
    }
  }
}

// ---------------------------------------------------------------------------
// relation-filtered segment-sum: one wave per edge.
// agg[dst] += tmp[src]  for edges with etype == rel   (128 f32 atomics/edge)
// ---------------------------------------------------------------------------
__global__ __launch_bounds__(256)
void scatter_rel(const float* __restrict__ tmp, const int* __restrict__ src,
                 const int* __restrict__ dst, const int* __restrict__ ety,
                 int rel, float* __restrict__ agg, int E) {
  long gid  = (long)blockIdx.x * blockDim.x + threadIdx.x;
  long edge = gid >> 5;
  int  lane = (int)(gid & 31);
  if (edge >= E) return;
  if (ety[edge] != rel) return;
  const float* ts = tmp + (size_t)src[edge] * DD;
  float*       ad = agg + (size_t)dst[edge] * DD;
#pragma unroll
  for (int t = 0; t < 4; ++t)
    atomicAdd(&ad[lane + 32 * t], ts[lane + 32 * t]);
}

// ---------------------------------------------------------------------------
// driver: for each layer l
//   agg = 0
//   for r in R:  tmp = h @ W[l,r] ; agg[dst] += tmp[src] where ety==r
//   h = agg + h @ loop_w[l] + bias[l]
// ---------------------------------------------------------------------------
extern "C" void kernel_launch(void* const* d_in, const int* in_sizes, int n_in,
                              void* d_out, int out_size, void* d_ws, size_t ws_size,
                              hipStream_t stream) {
  const float* h0   = (const float*)d_in[0];  // [N, D]
  const float* W    = (const float*)d_in[1];  // [L, R, D, D]
  const float* lw   = (const float*)d_in[2];  // [L, D, D]
  const float* bias = (const float*)d_in[3];  // [L, D]
  const int*   esrc = (const int*)d_in[4];    // [E]
  const int*   edst = (const int*)d_in[5];    // [E]
  const int*   ety  = (const int*)d_in[6];    // [E]

  const int Nn = in_sizes[0] / DD;
  const int E  = in_sizes[4];
  const int L  = in_sizes[2] / (DD * DD);
  const int R  = in_sizes[1] / (L * DD * DD);

  float* agg  = (float*)d_ws;
  float* tmp  = agg  + (size_t)Nn * DD;
  float* bufA = tmp  + (size_t)Nn * DD;
  float* bufB = bufA + (size_t)Nn * DD;
  float* out  = (float*)d_out;

  const long nd       = (long)Nn * DD;
  const int  zeroGrid = (int)((nd / 4 + 255) / 256);
  const int  gemmGrid = (Nn + 127) / 128;
  const int  scatGrid = (int)(((long)E * 32 + 255) / 256);

  const float* hin = h0;
  for (int l = 0; l < L; ++l) {
    float* hout = (l == L - 1) ? out : ((l & 1) ? bufB : bufA);

    zero_f32<<<zeroGrid, 256, 0, stream>>>(agg, nd);

    for (int r = 0; r < R; ++r) {
      const float* Wr = W + ((size_t)l * R + r) * DD * DD;
      gemm128_wmma<<<gemmGrid, 128, 0, stream>>>(hin, Wr, nullptr, nullptr, tmp, Nn);
      scatter_rel<<<scatGrid, 256, 0, stream>>>(tmp, esrc, edst, ety, r, agg, E);
    }

    gemm128_wmma<<<gemmGrid, 128, 0, stream>>>(
        hin, lw + (size_t)l * DD * DD, agg, bias + (size_t)l * DD, hout, Nn);

    hin = hout;
  }
}